// DecodeDetections_35639638622581
// MI455X (gfx1250) — compile-verified
//
#include <hip/hip_runtime.h>
#include <math.h>

#define B_SZ   16
#define N_BOX  8732
#define N_CLS  20        // classes 1..20 (conf channel cls+1)
#define K_NMS  200
#define TOPK   200
#define CONF_TH 0.01f
#define IOU_TH  0.45f
#define NEG_INF (-__builtin_inff())

// ---- NMS kernel LDS layout (dynamic shared, starts at LDS offset 0) ----
#define OFF_S    0                      // 8732 floats (scores)      34928 B
#define OFF_BOX  34944                  // 8732*4 floats (boxes)    139712 B
#define OFF_REDV 174656                 // 16 floats (per-wave max)
#define OFF_REDI 174720                 // 16 ints  (per-wave idx)
#define OFF_SELB 174784                 // 4 floats (selected box)
#define OFF_SELV 174800                 // 1 float  (selected val)
#define OFF_SELI 174804                 // 1 int    (selected idx)
#define NMS_SMEM 174816
#define NMS_THREADS 512
#define NMS_WAVES 16

// ---- TopK kernel LDS layout ----
#define TK_N      (N_CLS * K_NMS)       // 4000
#define TK_OFF_V  0                     // 4000 floats
#define TK_OFF_RV 16000                 // 8 floats
#define TK_OFF_RI 16032                 // 8 ints
#define TK_SMEM   16064
#define TK_THREADS 256
#define TK_WAVES   8

#if defined(__has_builtin)
#if __has_builtin(__builtin_amdgcn_tensor_load_to_lds)
#define HAS_TDM 1
#endif
#endif
#ifndef HAS_TDM
#define HAS_TDM 0
#endif

#if HAS_TDM
typedef unsigned int uint32x4 __attribute__((ext_vector_type(4)));
typedef int          int32x8  __attribute__((ext_vector_type(8)));
typedef int          int32x4  __attribute__((ext_vector_type(4)));

// Issue one 2D TDM load: tile (tile0 x tile1) of 4-byte elements, row stride
// stride0 (elements), from gaddr into LDS byte offset lds_off.
// D# packing per CDNA5 ISA §8.3/§8.4 (groups 2/3 padded with dims=1).
// clang-23 / therock-10.0 builtin arity: 6 args
//   (uint32x4 g0, int32x8 g1, int32x4 g2, int32x4 g3, int32x8 gx, i32 cpol)
__device__ __forceinline__ void tdm_load_2d_f32(unsigned lds_off,
                                                unsigned long long gaddr,
                                                unsigned tdim0, unsigned tdim1,
                                                unsigned tile0, unsigned tile1,
                                                unsigned long long stride0) {
  uint32x4 g0;
  g0[0] = 1u;                                               // count=1 (valid), user mode
  g0[1] = lds_off;                                          // lds_addr
  g0[2] = (unsigned)(gaddr & 0xFFFFFFFFull);                // global_addr[31:0]
  g0[3] = (unsigned)((gaddr >> 32) & 0x1FFFFFFull) | (2u << 30); // addr[56:32] | type=2
  int32x8 g1;
  g1[0] = (int)(2u << 16);                                  // data_size=4B; mask/flags=0
  g1[1] = (int)((tdim0 & 0xFFFFu) << 16);                   // tensor_dim0[15:0]
  g1[2] = (int)(((tdim0 >> 16) & 0xFFFFu) | ((tdim1 & 0xFFFFu) << 16));
  g1[3] = (int)(((tdim1 >> 16) & 0xFFFFu) | ((tile0 & 0xFFFFu) << 16));
  g1[4] = (int)(tile1 & 0xFFFFu);                           // tile_dim1 | tile_dim2=0
  g1[5] = (int)(stride0 & 0xFFFFFFFFull);                   // dim0_stride[31:0]
  g1[6] = (int)((stride0 >> 32) & 0xFFFFull);               // dim0_stride[47:32] | dim1_stride lo=0
  g1[7] = 0;
  int32x4 g2; g2[0] = 1; g2[1] = 1; g2[2] = 0; g2[3] = 0;   // tensor_dim2=1, tensor_dim3=1
  int32x4 g3; g3[0] = 0; g3[1] = (int)(1u << 16); g3[2] = 0; g3[3] = 0; // tensor_dim4=1
  int32x8 gx;                                               // unused trailing group
  gx[0] = 0; gx[1] = 0; gx[2] = 0; gx[3] = 0;
  gx[4] = 0; gx[5] = 0; gx[6] = 0; gx[7] = 0;
  __builtin_amdgcn_tensor_load_to_lds(g0, g1, g2, g3, gx, 0);
}
#endif

__device__ __forceinline__ void amax_combine(float& bv, int& bi, float ov, int oi) {
  if (ov > bv || (ov == bv && oi < bi)) { bv = ov; bi = oi; }
}

// ---------------- Kernel 1: box decode ----------------
__global__ void ssd_decode_kernel(const float* __restrict__ y,
                                  float* __restrict__ boxes, int total) {
  int i = blockIdx.x * blockDim.x + threadIdx.x;
  if (i >= total) return;
  const float* p = y + (size_t)i * 33;
  float l0 = p[21], l1 = p[22], l2 = p[23], l3 = p[24];
  float a0 = p[25], a1 = p[26], aw = p[27], ah = p[28];
  float v0 = p[29], v1 = p[30], v2 = p[31], v3 = p[32];
  float cx = l0 * v0 * aw + a0;
  float cy = l1 * v1 * ah + a1;
  float w  = expf(l2 * v2) * aw;
  float h  = expf(l3 * v3) * ah;
  float4 bb = make_float4(cx - 0.5f * w, cy - 0.5f * h, cx + 0.5f * w, cy + 0.5f * h);
  ((float4*)boxes)[i] = bb;
}

// ---------------- Kernel 2: per-(batch,class) greedy NMS ----------------
__global__ void ssd_nms_kernel(const float* __restrict__ y,
                               const float* __restrict__ boxes,
                               float* __restrict__ rows) {
  extern __shared__ char smem[];
  float* s      = (float*)(smem + OFF_S);
  float* bx     = (float*)(smem + OFF_BOX);
  float* redV   = (float*)(smem + OFF_REDV);
  int*   redI   = (int*)  (smem + OFF_REDI);
  float* selB   = (float*)(smem + OFF_SELB);
  float* selV   = (float*)(smem + OFF_SELV);
  int*   selI   = (int*)  (smem + OFF_SELI);

  const int tid = threadIdx.x;
  const int bc  = blockIdx.x;
  const int b   = bc / N_CLS;
  const int cls = bc % N_CLS;           // class id = cls+1, conf channel = cls+1

#if HAS_TDM
  if (tid < 32) {
    // Strided score-column gather: y[b, 0..N-1, cls+1], row stride 33 floats.
    unsigned long long sc_addr =
        (unsigned long long)(const void*)(y + ((size_t)b * N_BOX) * 33 + (cls + 1));
    tdm_load_2d_f32(OFF_S, sc_addr, 33u, (unsigned)N_BOX, 1u, (unsigned)N_BOX, 33ull);
    // Contiguous box tile: boxes[b, :, :] (N_BOX*4 floats, one row).
    unsigned long long bx_addr =
        (unsigned long long)(const void*)(boxes + (size_t)b * N_BOX * 4);
    tdm_load_2d_f32(OFF_BOX, bx_addr, (unsigned)(N_BOX * 4), 1u,
                    (unsigned)(N_BOX * 4), 1u, (unsigned long long)(N_BOX * 4));
    __builtin_amdgcn_s_wait_tensorcnt(0);
  }
  __syncthreads();
#else
  for (int j = tid; j < N_BOX; j += NMS_THREADS)
    s[j] = y[((size_t)b * N_BOX + j) * 33 + (cls + 1)];
  for (int j = tid; j < N_BOX * 4; j += NMS_THREADS)
    bx[j] = boxes[(size_t)b * N_BOX * 4 + j];
  __syncthreads();
#endif

  // Confidence mask (in LDS).
  for (int j = tid; j < N_BOX; j += NMS_THREADS)
    s[j] = (s[j] > CONF_TH) ? s[j] : NEG_INF;
  __syncthreads();

  float* rowOut = rows + ((size_t)bc) * K_NMS * 6;

  for (int it = 0; it < K_NMS; ++it) {
    // ---- argmax over s[0..N), tie -> smallest index (jnp.argmax semantics) ----
    float bv = NEG_INF; int bi = 0x7FFFFFFF;
    for (int j = tid; j < N_BOX; j += NMS_THREADS) {
      float v = s[j];
      if (v > bv) { bv = v; bi = j; }
    }
    for (int off = 16; off > 0; off >>= 1) {
      float ov = __shfl_down(bv, off, 32);
      int   oi = __shfl_down(bi, off, 32);
      amax_combine(bv, bi, ov, oi);
    }
    if ((tid & 31) == 0) { redV[tid >> 5] = bv; redI[tid >> 5] = bi; }
    __syncthreads();
    if (tid == 0) {
      float gv = redV[0]; int gi = redI[0];
      for (int w = 1; w < NMS_WAVES; ++w) amax_combine(gv, gi, redV[w], redI[w]);
      *selV = gv; *selI = gi;
      if (gv > NEG_INF) {
        float x1 = bx[4 * gi + 0], y1 = bx[4 * gi + 1];
        float x2 = bx[4 * gi + 2], y2 = bx[4 * gi + 3];
        selB[0] = x1; selB[1] = y1; selB[2] = x2; selB[3] = y2;
        rowOut[it * 6 + 0] = (float)(cls + 1);
        rowOut[it * 6 + 1] = gv;      // masked==raw when above threshold
        rowOut[it * 6 + 2] = x1;
        rowOut[it * 6 + 3] = y1;
        rowOut[it * 6 + 4] = x2;
        rowOut[it * 6 + 5] = y2;
      }
    }
    __syncthreads();

    float gv = *selV;
    if (!(gv > NEG_INF)) {
      // All remaining selections are invalid -> zero-fill remaining rows, done.
      for (int k = it * 6 + tid; k < K_NMS * 6; k += NMS_THREADS) rowOut[k] = 0.0f;
      break;                          // uniform across the workgroup
    }
    const int   gi = *selI;
    const float ax1 = selB[0], ay1 = selB[1], ax2 = selB[2], ay2 = selB[3];
    const float areaA = fmaxf(0.0f, ax2 - ax1) * fmaxf(0.0f, ay2 - ay1);

    for (int j = tid; j < N_BOX; j += NMS_THREADS) {
      if (s[j] == NEG_INF && j != gi) continue;   // already suppressed
      float x1 = bx[4 * j + 0], y1 = bx[4 * j + 1];
      float x2 = bx[4 * j + 2], y2 = bx[4 * j + 3];
      float iw = fmaxf(0.0f, fminf(ax2, x2) - fmaxf(ax1, x1));
      float ih = fmaxf(0.0f, fminf(ay2, y2) - fmaxf(ay1, y1));
      float inter = iw * ih;
      float areaB = fmaxf(0.0f, x2 - x1) * fmaxf(0.0f, y2 - y1);
      float uni = areaA + areaB - inter;
      float iou = (uni > 0.0f) ? (inter / uni) : 0.0f;
      if (iou > IOU_TH || j == gi) s[j] = NEG_INF;
    }
    __syncthreads();
  }
}

// ---------------- Kernel 3: per-batch top-200 (matches lax.top_k ordering) ----
__global__ void ssd_topk_kernel(const float* __restrict__ rows,
                                float* __restrict__ out) {
  extern __shared__ char smem[];
  float* v    = (float*)(smem + TK_OFF_V);
  float* redV = (float*)(smem + TK_OFF_RV);
  int*   redI = (int*)  (smem + TK_OFF_RI);
  const int tid = threadIdx.x;
  const int b   = blockIdx.x;
  const float* r = rows + (size_t)b * TK_N * 6;
  float* o = out + (size_t)b * TOPK * 6;

  for (int j = tid; j < TK_N; j += TK_THREADS) v[j] = r[j * 6 + 1];
  __syncthreads();

  for (int it = 0; it < TOPK; ++it) {
    float bv = NEG_INF; int bi = 0x7FFFFFFF;
    for (int j = tid; j < TK_N; j += TK_THREADS) {
      float x = v[j];
      if (x > bv) { bv = x; bi = j; }
    }
    for (int off = 16; off > 0; off >>= 1) {
      float ov = __shfl_down(bv, off, 32);
      int   oi = __shfl_down(bi, off, 32);
      amax_combine(bv, bi, ov, oi);
    }
    if ((tid & 31) == 0) { redV[tid >> 5] = bv; redI[tid >> 5] = bi; }
    __syncthreads();
    if (tid == 0) {
      float gv = redV[0]; int gi = redI[0];
      for (int w = 1; w < TK_WAVES; ++w) amax_combine(gv, gi, redV[w], redI[w]);
      for (int k = 0; k < 6; ++k) o[it * 6 + k] = r[gi * 6 + k];
      v[gi] = NEG_INF;
    }
    __syncthreads();
  }
}

extern "C" void kernel_launch(void* const* d_in, const int* in_sizes, int n_in,
                              void* d_out, int out_size, void* d_ws, size_t ws_size,
                              hipStream_t stream) {
  (void)in_sizes; (void)n_in; (void)out_size; (void)ws_size;
  const float* y = (const float*)d_in[0];
  float* boxes = (float*)d_ws;                                   // B*N*4 floats
  float* rows  = (float*)((char*)d_ws +
                          (size_t)B_SZ * N_BOX * 4 * sizeof(float)); // B*20*200*6
  float* out   = (float*)d_out;

  const int total = B_SZ * N_BOX;
  ssd_decode_kernel<<<(total + 255) / 256, 256, 0, stream>>>(y, boxes, total);
  ssd_nms_kernel<<<B_SZ * N_CLS, NMS_THREADS, NMS_SMEM, stream>>>(y, boxes, rows);
  ssd_topk_kernel<<<B_SZ, TK_THREADS, TK_SMEM, stream>>>(rows, out);
}